// RGB_Conv2d_19275813224500
// MI455X (gfx1250) — compile-verified
//
#include <hip/hip_runtime.h>
#include <hip/hip_bf16.h>
#include <math.h>

typedef __attribute__((ext_vector_type(2))) float v2f;
typedef __attribute__((ext_vector_type(8))) float v8f;

#define NB 16          // batch
#define NC 3           // input channels
#define NO 32          // output channels
#define HW 128         // input H=W
#define KW 5           // kernel
#define OHW 124        // output H=W
#define NP (OHW*OHW)   // 15376 pixels per plane (divisible by 16)
#define TILES_PER_IMG (NP/16)  // 961

// ---------------- Stage 1: 5x5 window sums of x and x^2 ----------------
__global__ void win_sums_kernel(const float* __restrict__ x,
                                float* __restrict__ s1,
                                float* __restrict__ s2) {
    int idx = blockIdx.x * blockDim.x + threadIdx.x;
    if (idx >= NB * NC * NP) return;
    int p  = idx % NP;
    int bc = idx / NP;
    int oh = p / OHW, ow = p % OHW;
    const float* xp = x + (size_t)bc * (HW * HW) + oh * HW + ow;
    float a1 = 0.0f, a2 = 0.0f;
#pragma unroll
    for (int i = 0; i < KW; ++i) {
#pragma unroll
        for (int j = 0; j < KW; ++j) {
            float v = xp[i * HW + j];
            a1 += v;
            a2 += v * v;
        }
    }
    s1[idx] = a1;
    s2[idx] = a2;
}

// ---------------- Stage 2: dist via WMMA f32 16x16x4 ----------------
// inner[m][n] = s2[m]*1 + s1[m]*(-2*w_n) + 1*(25*w_n^2)
//   A (16x4): row m = [s2, s1, 1, 0]     (lanes 0-15: K=0,1 ; lanes 16-31: K=2,3)
//   B (4x16): col n = [1, -2w, 25w^2, 0] (lanes 0-15: K=0,1 ; lanes 16-31: K=2,3)
// dist[b,o,pix] = sum_c sqrt(max(inner,0)); stored directly in out layout (b,o,p).
__global__ void dist_wmma_kernel(const float* __restrict__ s1,
                                 const float* __restrict__ s2,
                                 const float* __restrict__ wgt,   // rgb_weight (32,3)
                                 float* __restrict__ dist_out) {
    int wave = (blockIdx.x * blockDim.x + threadIdx.x) >> 5;
    int lane = threadIdx.x & 31;
    int b  = wave / TILES_PER_IMG;
    int t  = wave % TILES_PER_IMG;
    int p0 = t * 16;
    int m  = lane & 15;
    bool hi = lane >= 16;

    v8f acc0 = {};  // channels 0..15
    v8f acc1 = {};  // channels 16..31
#pragma unroll
    for (int c = 0; c < NC; ++c) {
        v2f A;
        if (!hi) {
            int si = (b * NC + c) * NP + p0 + m;   // pixel row m of the tile
            A.x = s2[si];   // K=0
            A.y = s1[si];   // K=1
        } else {
            A.x = 1.0f;     // K=2 (constant row -> picks up 25*w^2)
            A.y = 0.0f;     // K=3
        }
        // wT[c][o] = rgb_weight[o*3 + c]
        float w0 = wgt[(m)      * NC + c];   // o = m        (half 0)
        float w1 = wgt[(16 + m) * NC + c];   // o = 16 + m   (half 1)
        v2f B0, B1;
        if (!hi) {
            B0.x = 1.0f;           B0.y = -2.0f * w0;   // K=0, K=1
            B1.x = 1.0f;           B1.y = -2.0f * w1;
        } else {
            B0.x = 25.0f * w0 * w0; B0.y = 0.0f;        // K=2, K=3
            B1.x = 25.0f * w1 * w1; B1.y = 0.0f;
        }
        v8f z = {};
        v8f i0 = __builtin_amdgcn_wmma_f32_16x16x4_f32(
            false, A, false, B0, (short)0, z, false, false);
        v8f i1 = __builtin_amdgcn_wmma_f32_16x16x4_f32(
            false, A, false, B1, (short)0, z, false, false);
#pragma unroll
        for (int v = 0; v < 8; ++v) {
            acc0[v] += sqrtf(fmaxf(i0[v], 0.0f));
            acc1[v] += sqrtf(fmaxf(i1[v], 0.0f));
        }
    }
    // D layout: VGPR v, lanes 0-15 -> M=v ; lanes 16-31 -> M=v+8 ; N = lane&15
    int pbase = p0 + (hi ? 8 : 0);
#pragma unroll
    for (int v = 0; v < 8; ++v) {
        int pix = pbase + v;
        dist_out[((size_t)(b * NO + m)      * NP) + pix] = acc0[v];
        dist_out[((size_t)(b * NO + 16 + m) * NP) + pix] = acc1[v];
    }
}

// ---------------- Stage 3: per-pixel std over (B,O) + gaussian RBF, in place ----------------
__global__ void stats_rbf_kernel(float* __restrict__ out) {
    int wave = (blockIdx.x * blockDim.x + threadIdx.x) >> 5;
    int lane = threadIdx.x & 31;           // lane == output channel o
    int p = wave;                          // pixel position
    if (p >= NP) return;                   // wave-uniform

    float d[NB];
    float sum = 0.0f, sq = 0.0f;
#pragma unroll
    for (int b = 0; b < NB; ++b) {
        float v = out[((size_t)(b * NO + lane) * NP) + p];
        d[b] = v;
        sum += v;
        sq  += v * v;
    }
#pragma unroll
    for (int off = 16; off >= 1; off >>= 1) {
        sum += __shfl_xor(sum, off, 32);
        sq  += __shfl_xor(sq,  off, 32);
    }
    const float n = (float)(NB * NO);                    // 512
    float var = (sq - sum * sum / n) / (n - 1.0f);       // unbiased
    var = fmaxf(var, 0.0f);
    float istd = rsqrtf(var);                            // var==0 -> inf -> exp(-inf)=0
#pragma unroll
    for (int b = 0; b < NB; ++b) {
        float tdd = d[b] * istd;
        out[((size_t)(b * NO + lane) * NP) + p] = expf(-0.5f * tdd * tdd);
    }
}

extern "C" void kernel_launch(void* const* d_in, const int* in_sizes, int n_in,
                              void* d_out, int out_size, void* d_ws, size_t ws_size,
                              hipStream_t stream) {
    const float* x   = (const float*)d_in[0];   // (16,3,128,128)
    const float* wgt = (const float*)d_in[1];   // (32,3)
    float* out = (float*)d_out;                 // (16,32,124,124)

    float* s1 = (float*)d_ws;                   // (16,3,15376)
    float* s2 = s1 + (size_t)NB * NC * NP;      // (16,3,15376)  total ws ~5.9 MB

    // Stage 1: window sums
    {
        int n = NB * NC * NP;                   // 738,048
        win_sums_kernel<<<(n + 255) / 256, 256, 0, stream>>>(x, s1, s2);
    }
    // Stage 2: WMMA distance, one wave per 16-pixel tile
    {
        int waves = NB * TILES_PER_IMG;         // 15,376
        int threads = waves * 32;               // 492,032
        dist_wmma_kernel<<<threads / 256, 256, 0, stream>>>(s1, s2, wgt, out);
    }
    // Stage 3: per-pixel std + RBF, one wave per pixel, in place on d_out
    {
        int blocks = NP / 8;                    // 1922 (8 waves of 32 per block)
        stats_rbf_kernel<<<blocks, 256, 0, stream>>>(out);
    }
}